// FreTransfer_28441273434572
// MI455X (gfx1250) — compile-verified
//
#include <hip/hip_runtime.h>
#include <stdint.h>

// 2-D Haar analysis DWT, stride-2 2x2 separable filter bank.
// x:   (8, 64, 512, 512) fp32
// out: (8, 256, 256, 256) fp32, out[b, i*64+c] = filter_i applied to x[b, c]
//
// Memory-bound: 1 GiB total traffic -> ~46us floor at 23.3 TB/s.
// Data path: async global->LDS (ASYNCcnt ops), b128 LDS reads,
// non-temporal b64 stores (read-once/write-once; don't pollute L2).

typedef float v2f __attribute__((ext_vector_type(2)));
typedef float v4f __attribute__((ext_vector_type(4)));
typedef int   v4i __attribute__((ext_vector_type(4)));

#define BATCH 8
#define CH    64
#define IN_H  512
#define IN_W  512
#define OUT_H 256
#define OUT_W 256

typedef __attribute__((address_space(3))) char lchar;

__global__ __launch_bounds__(256)
void haar_dwt2x2(const float* __restrict__ x,
                 const float* __restrict__ w1,
                 const float* __restrict__ w2,
                 float* __restrict__ out)
{
    // Filter taps: uniform across the grid -> scalar loads.
    const float h00 = w1[0], h01 = w1[1];   // lowpass row taps
    const float h10 = w2[0], h11 = w2[1];   // highpass row taps

    const int t     = threadIdx.x;          // 0..255
    const int blk   = blockIdx.x;           // 0..65535
    const int ypair = blk & 127;            // which pair of output rows
    const int bc    = blk >> 7;             // b*CH + c

    // 4 consecutive input rows feed this block's 2 output rows.
    const float* src = x + ((size_t)bc * IN_H + (size_t)ypair * 4) * IN_W;

    __shared__ __align__(16) float lds[4 * IN_W];   // 8 KB

    // ---- stage 4 input rows into LDS via async data mover ----
    {
        char* g = (char*)const_cast<float*>(src) + t * 16;
        char* l = (char*)lds + t * 16;
#if __has_builtin(__builtin_amdgcn_global_load_async_to_lds_b128)
        // Builtin signature (from hipcc diagnostic): (v4i*, v4i*, imm, imm).
        // INST_OFFSET applies to both LDS and global address (ISA 15.18.3),
        // so the same pointers cover bytes [0,4096) and [4096,8192).
        __builtin_amdgcn_global_load_async_to_lds_b128((v4i*)g, (v4i*)l, 0, 0);
        __builtin_amdgcn_global_load_async_to_lds_b128((v4i*)g, (v4i*)l, 4096, 0);
#else
        unsigned loff = (unsigned)(uintptr_t)(lchar*)l;   // LDS byte offset
        asm volatile("global_load_async_to_lds_b128 %0, %1, off"
                     :: "v"(loff), "v"(g) : "memory");
        asm volatile("global_load_async_to_lds_b128 %0, %1, off offset:4096"
                     :: "v"(loff), "v"(g) : "memory");
#endif
    }
#if __has_builtin(__builtin_amdgcn_s_wait_asynccnt)
    __builtin_amdgcn_s_wait_asynccnt(0);
#else
    asm volatile("s_wait_asynccnt 0x0" ::: "memory");
#endif
    __syncthreads();

    // ---- compute: each thread produces 2 output pixels of one output row ----
    const int r  = t >> 7;       // output row within the pair (0/1)
    const int xp = t & 127;      // output x-pair -> x = 2*xp, 2*xp+1

    const v4f top = *(const v4f*)&lds[(2 * r    ) * IN_W + 4 * xp];
    const v4f bot = *(const v4f*)&lds[(2 * r + 1) * IN_W + 4 * xp];

    // Row filtering (separable): P = lowpass, Q = highpass.
    const float Pt0 = h00 * top.x + h01 * top.y;
    const float Pt1 = h00 * top.z + h01 * top.w;
    const float Qt0 = h10 * top.x + h11 * top.y;
    const float Qt1 = h10 * top.z + h11 * top.w;
    const float Pb0 = h00 * bot.x + h01 * bot.y;
    const float Pb1 = h00 * bot.z + h01 * bot.w;
    const float Qb0 = h10 * bot.x + h11 * bot.y;
    const float Qb1 = h10 * bot.z + h11 * bot.w;

    // Column filtering. Reference filter order: [h0h0t, h0h1t, h1h0t, h1h1t].
    v2f o0 = { h00 * Pt0 + h01 * Pb0, h00 * Pt1 + h01 * Pb1 };  // LL
    v2f o1 = { h10 * Pt0 + h11 * Pb0, h10 * Pt1 + h11 * Pb1 };  // row-low, col-high
    v2f o2 = { h00 * Qt0 + h01 * Qb0, h00 * Qt1 + h01 * Qb1 };  // row-high, col-low
    v2f o3 = { h10 * Qt0 + h11 * Qb0, h10 * Qt1 + h11 * Qb1 };  // HH

    const int b = bc >> 6;       // / CH
    const int c = bc & 63;       // % CH
    const int y = ypair * 2 + r;
    float* base = out + (((size_t)(b * 4 * CH + c)) * OUT_H + y) * OUT_W + 2 * xp;
    const size_t ps = (size_t)CH * OUT_H * OUT_W;   // stride between filter planes

    __builtin_nontemporal_store(o0, (v2f*)(base));
    __builtin_nontemporal_store(o1, (v2f*)(base + ps));
    __builtin_nontemporal_store(o2, (v2f*)(base + 2 * ps));
    __builtin_nontemporal_store(o3, (v2f*)(base + 3 * ps));
}

extern "C" void kernel_launch(void* const* d_in, const int* in_sizes, int n_in,
                              void* d_out, int out_size, void* d_ws, size_t ws_size,
                              hipStream_t stream) {
    (void)in_sizes; (void)n_in; (void)out_size; (void)d_ws; (void)ws_size;
    const float* x  = (const float*)d_in[0];
    const float* w1 = (const float*)d_in[1];
    const float* w2 = (const float*)d_in[2];
    float* out = (float*)d_out;

    dim3 grid(BATCH * CH * (OUT_H / 2));   // 65536 blocks
    dim3 block(256);                       // 8 wave32 per block
    haar_dwt2x2<<<grid, block, 0, stream>>>(x, w1, w2, out);
}